// PBC_23201413333269
// MI455X (gfx1250) — compile-verified
//
#include <hip/hip_runtime.h>
#include <stdint.h>

// ---------------------------------------------------------------------------
// PBC nonlinear-compensation triple-product stencil, MI455X (gfx1250, wave32)
//   acc[k,i] = sum_t c_t * s[k,t] * E[k-m_t, i]
//   s[k,t]   = sum_i E[k-n_t,i] * conj(E[k-m_t-n_t,i])   (shared by both modes)
// VALU-bound (≈2.1 GFLOP fp32, ≈4 MB traffic) -> LDS-tiled stencil.
// CDNA5 paths: global_load_async_to_lds_b32 staging (ASYNCcnt) + s_wait_asynccnt,
// global_prefetch_b8. 2 k's per thread; one b128 term-table load per term.
// ---------------------------------------------------------------------------

#ifndef __has_builtin
#define __has_builtin(x) 0
#endif

#if __has_builtin(__builtin_amdgcn_global_load_async_to_lds_b32)
#define PBC_HAVE_ASYNC_LDS 1
#else
#define PBC_HAVE_ASYNC_LDS 0
#endif

#define S_TOT   16384
#define L_OVL   50
#define T_TERMS 449
#define NTHREADS 256
#define KPT      2                          // k's per thread
#define TILE_K  (NTHREADS * KPT)            // 512 output k's per block
#define HALO    50                          // max |shift| (|m|,|n|,|m+n| <= 50)
#define EW      (TILE_K + 2 * HALO)         // 612 staged E entries per tile
#define K_LO    L_OVL                       // 50
#define K_HI    (S_TOT - L_OVL)             // 16334
#define K_OUT   (K_HI - K_LO)               // 16284

typedef __attribute__((address_space(1))) int as1_int;
typedef __attribute__((address_space(3))) int as3_int;

// Exclusive prefix sums of row term-counts cnt(m) = #{n in [-25,25] : |m*n|<=25}
// for m = -25..25 (cnt: 3 x13, 5 x4, 7 x2, 9, 11, 13, 17, 25, 51, 51, 51, ... mirrored).
__constant__ unsigned short g_rowbase[51] = {
      0,   3,   6,   9,  12,  15,  18,  21,  24,  27,  30,  33,  36, // m=-25..-13
     39,  44,  49,  54,                                               // m=-12..-9
     59,  66,                                                         // m=-8,-7
     73,  82,  93, 106, 123, 148,                                     // m=-6..-1
    199, 250,                                                         // m=0,1
    301, 326, 343, 356, 367, 376, 383,                                // m=2..8
    390, 395, 400, 405,                                               // m=9..12 (base)
    410, 413, 416, 419, 422, 425, 428, 431, 434, 437, 440, 443, 446  // m=13..25
};
// (index i = m+25; g_rowbase[50] = 446, last row has 3 terms -> total 449)

__global__ __launch_bounds__(NTHREADS) void pbc_kernel(
        const float* __restrict__ Ere, const float* __restrict__ Eim,
        const float* __restrict__ Cre, const float* __restrict__ Cim,
        float4* __restrict__ out)
{
    __shared__ float4 s_E[EW];          // per k: {re0, im0, re1, im1} -> one b128
    __shared__ float4 s_term[T_TERMS];  // {c_re, c_im, bitcast(m | n<<16), 0}

    const int tid  = threadIdx.x;
    const int tile = blockIdx.x;
    const int b    = blockIdx.y;
    const int g0   = tile * TILE_K;     // first global E index staged for this tile

    // gfx1250 prefetch (global_prefetch_b8) of the coefficient rows
    __builtin_prefetch(Cre + tid, 0, 0);
    __builtin_prefetch(Cim + tid, 0, 0);

    // --- build (m, n, c) term table in LDS, same order as the reference -----
    if (tid < 51) {
        const int m = tid - 25;
        int p = (int)g_rowbase[tid];
        for (int n = -25; n <= 25; ++n) {
            int prod = m * n; prod = prod < 0 ? -prod : prod;
            if (prod <= 25) {
                const int v = (m & 0xFFFF) | (n << 16);
                s_term[p] = make_float4(Cre[p], Cim[p], __int_as_float(v), 0.0f);
                ++p;
            }
        }
    }

    // --- stage E window into LDS (interleaved re0,im0,re1,im1 per k) --------
    // word sub: 0 -> re mode0, 1 -> im mode0, 2 -> re mode1, 3 -> im mode1
    const size_t ebase = (size_t)b * (size_t)S_TOT * 2u;
    float* sEw = (float*)s_E;
    for (int w = tid; w < EW * 4; w += NTHREADS) {
        const int j   = w >> 2;
        const int sub = w & 3;
        const int g   = g0 + j;
        if (g < S_TOT) {
            const float* src = ((sub & 1) ? Eim : Ere) + ebase + (size_t)g * 2u + (sub >> 1);
#if PBC_HAVE_ASYNC_LDS
            __builtin_amdgcn_global_load_async_to_lds_b32(
                (as1_int*)(int*)(float*)src, (as3_int*)(sEw + w), 0, 0);
#else
            sEw[w] = *src;
#endif
        }
    }
#if PBC_HAVE_ASYNC_LDS
#if __has_builtin(__builtin_amdgcn_s_wait_asynccnt)
    __builtin_amdgcn_s_wait_asynccnt(0);
#else
    asm volatile("s_wait_asynccnt 0" ::: "memory");
#endif
#endif
    __syncthreads();

    // --- each thread accumulates 2 k's: kl0 = HALO+tid, kl1 = kl0+NTHREADS --
    const int kl0 = HALO + tid;
    const int kl1 = kl0 + NTHREADS;
    float p0r = 0.f, p0i = 0.f, p1r = 0.f, p1i = 0.f;   // k0: mode0, mode1
    float q0r = 0.f, q0i = 0.f, q1r = 0.f, q1i = 0.f;   // k1: mode0, mode1

    #pragma unroll 2
    for (int t = 0; t < T_TERMS; ++t) {
        const float4 tc = s_term[t];                 // one ds_load_b128
        const int v  = __float_as_int(tc.z);
        const int m  = (int)(short)v;
        const int n  = v >> 16;
        const float cr = tc.x, ci = tc.y;
        {   // ---- k0 ----
            const float4 en  = s_E[kl0 - n];         // E[k-n]
            const float4 emn = s_E[kl0 - m - n];     // E[k-m-n]
            const float4 em  = s_E[kl0 - m];         // E[k-m]
            const float sre = en.x*emn.x + en.y*emn.y + en.z*emn.z + en.w*emn.w;
            const float sim = en.y*emn.x - en.x*emn.y + en.w*emn.z - en.z*emn.w;
            const float scr = sre*cr - sim*ci;
            const float sci = sre*ci + sim*cr;
            p0r += scr*em.x - sci*em.y;
            p0i += scr*em.y + sci*em.x;
            p1r += scr*em.z - sci*em.w;
            p1i += scr*em.w + sci*em.z;
        }
        {   // ---- k1 ----
            const float4 en  = s_E[kl1 - n];
            const float4 emn = s_E[kl1 - m - n];
            const float4 em  = s_E[kl1 - m];
            const float sre = en.x*emn.x + en.y*emn.y + en.z*emn.z + en.w*emn.w;
            const float sim = en.y*emn.x - en.x*emn.y + en.w*emn.z - en.z*emn.w;
            const float scr = sre*cr - sim*ci;
            const float sci = sre*ci + sim*cr;
            q0r += scr*em.x - sci*em.y;
            q0i += scr*em.y + sci*em.x;
            q1r += scr*em.z - sci*em.w;
            q1i += scr*em.w + sci*em.z;
        }
    }

    // --- add E[k] and store (complex64 x 2 modes = one float4 per k) --------
    const int k0 = K_LO + g0 + tid;
    if (k0 < K_HI) {
        const float4 e = s_E[kl0];
        out[(size_t)b * (size_t)K_OUT + (size_t)(k0 - K_LO)] =
            make_float4(e.x + p0r, e.y + p0i, e.z + p1r, e.w + p1i);
    }
    const int k1 = k0 + NTHREADS;
    if (k1 < K_HI) {
        const float4 e = s_E[kl1];
        out[(size_t)b * (size_t)K_OUT + (size_t)(k1 - K_LO)] =
            make_float4(e.x + q0r, e.y + q0i, e.z + q1r, e.w + q1i);
    }
}

extern "C" void kernel_launch(void* const* d_in, const int* in_sizes, int n_in,
                              void* d_out, int out_size, void* d_ws, size_t ws_size,
                              hipStream_t stream) {
    const float* Ere = (const float*)d_in[0];   // [B, S, 2] f32
    const float* Eim = (const float*)d_in[1];   // [B, S, 2] f32
    const float* Cre = (const float*)d_in[2];   // [4, 449] f32 (row 0 used)
    const float* Cim = (const float*)d_in[3];   // [4, 449] f32 (row 0 used)

    const int B = in_sizes[0] / (S_TOT * 2);    // 8 for the reference setup
    dim3 grid((K_OUT + TILE_K - 1) / TILE_K, B > 0 ? B : 1);
    pbc_kernel<<<grid, NTHREADS, 0, stream>>>(Ere, Eim, Cre, Cim, (float4*)d_out);
}